// CTCLossWrapper_81003083202600
// MI455X (gfx1250) — compile-verified
//
#include <hip/hip_runtime.h>

#define CTC_NEG -1e30f

__device__ __forceinline__ float lae2(float a, float b) {
    float m = fmaxf(a, b);
    return m + log1pf(expf(-fabsf(a - b)));
}

__device__ __forceinline__ float wred_max(float v) {
#pragma unroll
    for (int m = 16; m > 0; m >>= 1) v = fmaxf(v, __shfl_xor(v, m, 32));
    return v;
}
__device__ __forceinline__ float wred_sum(float v) {
#pragma unroll
    for (int m = 16; m > 0; m >>= 1) v += __shfl_xor(v, m, 32);
    return v;
}

// logsumexp over 256 floats sitting in LDS; executed by one wave (32 lanes)
__device__ __forceinline__ float lse256(const float* rb, int lane) {
    float x[8];
#pragma unroll
    for (int k = 0; k < 8; ++k) x[k] = rb[lane + 32 * k];
    float m = x[0];
#pragma unroll
    for (int k = 1; k < 8; ++k) m = fmaxf(m, x[k]);
    m = wred_max(m);
    float s = 0.f;
#pragma unroll
    for (int k = 0; k < 8; ++k) s += expf(x[k] - m);
    s = wred_sum(s);
    return m + logf(s);
}

// Low 32 bits of a generic pointer to LDS == byte offset within the wave's
// LDS allocation (flat aperture keeps the offset in addr[31:0]); this is the
// VDST operand the async-DMA expects.
__device__ __forceinline__ unsigned lds_off(const void* p) {
    return (unsigned)(unsigned long long)p;
}

// Async-DMA one 256-float row (1 KB) global -> LDS: two b128 ops per lane,
// wave32. Tracked by ASYNCcnt (2 increments per row).
__device__ __forceinline__ void async_row(const float* src, float* dst, int lane) {
    unsigned l0 = lds_off(dst) + (unsigned)(lane << 4);
    unsigned v0 = (unsigned)(lane << 4);
    asm volatile("global_load_async_to_lds_b128 %0, %1, %2"
                 :: "v"(l0), "v"(v0), "s"(src) : "memory");
    asm volatile("global_load_async_to_lds_b128 %0, %1, %2 offset:512"
                 :: "v"(l0), "v"(v0), "s"(src) : "memory");
}

// Fused log_softmax + CTC forward DP. One workgroup per batch element.
// LDS layout: rowbuf[4][V] | A0[L+3] | A1[L+3] | lse[2]
__global__ void ctc_fused(const float* __restrict__ lp,   // [T,B,V]
                          const int* __restrict__ targets,// [B,S]
                          const int* __restrict__ in_len, // [B]
                          const int* __restrict__ tg_len, // [B]
                          float* __restrict__ loss_out,   // [B]
                          int T, int B, int S) {
    constexpr int V = 256;
    const int b   = blockIdx.x;
    const int tid = threadIdx.x;
    const int L   = 2 * S + 1;

    extern __shared__ float smem[];
    float* rowb = smem;                 // 4 * V floats (16B aligned)
    float* A0   = smem + 4 * V;         // L+3
    float* A1   = A0 + (L + 3);         // L+3
    float* lseS = A1 + (L + 3);         // 2

    const int inLen = in_len[b];
    const int tl    = tg_len[b];
    const size_t rowStride = (size_t)B * V;
    const float* lpb = lp + (size_t)b * V;   // row r base = lpb + r*rowStride

    // ---- per-thread DP constants: ext[l] and skip[l] ----
    int  gidx = 0;
    bool skip = false;
    if (tid < L && (tid & 1)) {
        int j  = (tid - 1) >> 1;
        int tg = targets[(size_t)b * S + j];
        int pv = j ? targets[(size_t)b * S + j - 1] : -1;
        gidx = tg;
        skip = (tg != 0) && (tg != pv);
    }

    // ---- preload rows 0 and 1 (synchronous) ----
    if (tid < V) {
        rowb[tid] = lpb[tid];
        if (T > 1) rowb[V + tid] = lpb[rowStride + tid];
    }
    __syncthreads();

    if (tid < 32) {
        float l0 = lse256(rowb, tid);
        float l1 = (T > 1) ? lse256(rowb + V, tid) : 0.f;
        if (tid == 0) { lseS[0] = l0; lseS[1] = l1; }
    }
    __syncthreads();

    // ---- alpha(t=0) ----
    if (tid < L) A0[2 + tid] = CTC_NEG;
    if (tid == 0) {
        A0[0] = A0[1] = CTC_NEG;
        A1[0] = A1[1] = CTC_NEG;
        A0[2] = rowb[0] - lseS[0];
    }
    if (tid == 1 && L > 1) A0[3] = rowb[gidx] - lseS[0];
    __syncthreads();

    // ---- start the DMA pipeline (rows 2 and 3) ----
    if (tid < 32) {
        if (2 < T) async_row(lpb + 2 * rowStride, rowb + 2 * V, tid);
        if (3 < T) async_row(lpb + 3 * rowStride, rowb + 3 * V, tid);
    }

    float* Aprev = A0;
    float* Anew  = A1;

    for (int t = 1; t < T; ++t) {
        const float* rc = rowb + (size_t)(t & 3) * V;

        if (tid < 32) {
            // Make row t+1 resident: rows t+1,t+2 may be in flight (<=4 async
            // ops); async loads retire in order, so cnt<=2 retires row t+1.
            if (t + 2 < T) asm volatile("s_wait_asynccnt 2" ::: "memory");
            else           asm volatile("s_wait_asynccnt 0" ::: "memory");
            if (t + 1 < T) {
                float lv = lse256(rowb + ((t + 1) & 3) * V, tid);
                if (tid == 0) lseS[(t + 1) & 1] = lv;
            }
        }

        if (tid < L) {
            float a   = Aprev[2 + tid];
            float a1v = Aprev[1 + tid];
            float a2v = Aprev[tid];
            float comb = lae2(a, a1v);
            if (skip) comb = lae2(comb, a2v);
            float lpv = rc[gidx] - lseS[t & 1];
            float nv  = comb + lpv;
            if (t >= inLen) nv = a;
            Anew[2 + tid] = nv;
        }
        __syncthreads();

        if (tid < 32 && (t + 3) < T)
            async_row(lpb + (size_t)(t + 3) * rowStride, rowb + ((t + 3) & 3) * V, tid);
        if (tid < 8 && (t + 8) < T)   // warm L2 ahead of the DMA engine
            __builtin_prefetch((const char*)(lpb + (size_t)(t + 8) * rowStride) + tid * 128, 0, 1);

        float* tmp = Aprev; Aprev = Anew; Anew = tmp;
    }

    if (tid == 0) {
        float e1 = Aprev[2 + 2 * tl];
        float e2 = Aprev[2 + 2 * tl - 1];
        float lb = -lae2(e1, e2);
        if (!(isfinite(lb) && lb < 1e29f)) lb = 0.f;   // zero_infinity
        loss_out[b] = lb / (float)tl;
    }
}

// Deterministic shuffle-tree mean of B per-batch losses.
__global__ void mean_reduce(const float* __restrict__ v, float* __restrict__ out, int n) {
    int lane = threadIdx.x;
    float s = 0.f;
    for (int i = lane; i < n; i += 32) s += v[i];
#pragma unroll
    for (int m = 16; m > 0; m >>= 1) s += __shfl_xor(s, m, 32);
    if (lane == 0) out[0] = s / (float)n;
}

extern "C" void kernel_launch(void* const* d_in, const int* in_sizes, int n_in,
                              void* d_out, int out_size, void* d_ws, size_t ws_size,
                              hipStream_t stream) {
    const float* lp      = (const float*)d_in[0];
    const int*   targets = (const int*)d_in[1];
    const int*   inl     = (const int*)d_in[2];
    const int*   tgl     = (const int*)d_in[3];

    const int V = 256;
    const int B = in_sizes[2];
    const int S = in_sizes[1] / B;
    const int T = in_sizes[0] / (B * V);
    const int L = 2 * S + 1;

    int NT = ((L + 31) / 32) * 32;
    if (NT < 256) NT = 256;                       // preamble needs 256 loaders

    size_t smem_bytes = (size_t)(4 * V + 2 * (L + 3) + 4) * sizeof(float);

    float* ws = (float*)d_ws;                     // B floats of scratch
    ctc_fused<<<B, NT, smem_bytes, stream>>>(lp, targets, inl, tgl, ws, T, B, S);
    mean_reduce<<<1, 32, 0, stream>>>(ws, (float*)d_out, B);
}